// DeepseekV2MoE_28613072126251
// MI455X (gfx1250) — compile-verified
//
#include <hip/hip_runtime.h>
#include <hip/hip_bf16.h>

#define HID      2048
#define FFN_DIM  1408
#define NEXP     64
#define TOPK     6
#define SFFN     2816     // shared FFN = 2*1408
#define NTOK     256
#define LDS_STR  1412     // fused-fallback act stride (floats)
#define ROUTED_ROWS 3584  // worst-case chunk-padded routed act rows (1536+64*31 <= 3584)
#define ACT_ROWS 4096     // + 512 shared rows

typedef __attribute__((ext_vector_type(2))) float v2f;
typedef __attribute__((ext_vector_type(8))) float v8f;

__device__ __forceinline__ v8f wmma_f32_4(v2f a, v2f b, v8f c) {
    // D = A(16x4 f32) x B(4x16 f32) + C(16x16 f32)
    return __builtin_amdgcn_wmma_f32_16x16x4_f32(
        false, a, false, b, (short)0, c, false, false);
}

__device__ __forceinline__ float silu_fast(float g) {
    // v_rcp_f32 approx; denominator >= 1 so well conditioned
    return g * __builtin_amdgcn_rcpf(1.f + __expf(-g));
}

// ---------------------------------------------------------------------------
// Kernel 1: routing. One block per token, 64 threads (one per expert).
// ---------------------------------------------------------------------------
__global__ __launch_bounds__(64) void moe_route_kernel(
    const float* __restrict__ x,        // [256][2048]
    const float* __restrict__ gate_w,   // [64][2048]
    int*   __restrict__ counts,         // [64], pre-zeroed
    int*   __restrict__ etok,           // [64][256]
    float* __restrict__ ewt)            // [64][256]
{
    const int t = blockIdx.x;
    const int e = threadIdx.x;

    const float* xr = x + (size_t)t * HID;
    const float* wr = gate_w + (size_t)e * HID;
    float acc = 0.f;
    #pragma unroll 4
    for (int k = 0; k < HID; k += 4) {
        float4 xv = *(const float4*)(xr + k);
        float4 wv = *(const float4*)(wr + k);
        acc += xv.x * wv.x + xv.y * wv.y + xv.z * wv.z + xv.w * wv.w;
    }

    __shared__ float slog[NEXP];
    __shared__ float sprob[NEXP];
    __shared__ float smax;
    slog[e] = acc;
    __syncthreads();
    if (e == 0) {
        float mx = slog[0];
        for (int i = 1; i < NEXP; ++i) mx = fmaxf(mx, slog[i]);
        smax = mx;
    }
    __syncthreads();
    sprob[e] = __expf(slog[e] - smax);
    __syncthreads();

    if (e == 0) {
        float denom = 0.f;
        for (int i = 0; i < NEXP; ++i) denom += sprob[i];
        float inv = 1.f / denom;
        int   ids[TOPK];
        float wts[TOPK];
        float tsum = 0.f;
        for (int j = 0; j < TOPK; ++j) {
            int bi = 0; float bv = -1.f;
            for (int i = 0; i < NEXP; ++i) {
                float v = sprob[i];
                if (v > bv) { bv = v; bi = i; }
            }
            sprob[bi] = -2.f;
            ids[j] = bi;
            wts[j] = bv * inv;
            tsum  += bv * inv;
        }
        float norm = 1.f / (tsum + 1e-20f);
        for (int j = 0; j < TOPK; ++j) {
            int ex = ids[j];
            int slot = atomicAdd(&counts[ex], 1);
            etok[ex * NTOK + slot] = t;
            ewt [ex * NTOK + slot] = wts[j] * norm;
        }
    }
}

// ---------------------------------------------------------------------------
// Kernel 1b: chunk-aligned (32-token) exclusive prefix over expert counts ->
// compact activation-row offsets.
// ---------------------------------------------------------------------------
__global__ void moe_offsets_kernel(const int* __restrict__ counts,
                                   int* __restrict__ offsets)
{
    if (threadIdx.x == 0 && blockIdx.x == 0) {
        int off = 0;
        for (int e = 0; e < NEXP; ++e) {
            offsets[e] = off;
            off += ((counts[e] + 31) >> 5) << 5;
        }
    }
}

// ---------------------------------------------------------------------------
// Kernel 2: gate/up GEMM + silu, M=32 per chunk (B fragments reused across
// two M-tiles -> expert weights streamed once per 32-token chunk).
// grid = (66 experts, 8 chunks, 11 f-splits of 128 cols); block = 8 waves.
// ---------------------------------------------------------------------------
__global__ __launch_bounds__(256) void moe_gateup_kernel(
    const float* __restrict__ x,           // [256][2048]
    const float* __restrict__ w_gate_up,   // [64][2*1408][2048]
    const float* __restrict__ ws_gate_up,  // [2*2816][2048]
    const int*   __restrict__ counts,
    const int*   __restrict__ etok,
    const float* __restrict__ ewt,
    const int*   __restrict__ offsets,
    float* __restrict__ actws)             // [4096][1408]
{
    __shared__ int   stok[32];
    __shared__ float swt[32];

    const int e      = blockIdx.x;
    const int chunk  = blockIdx.y;
    const int fsplit = blockIdx.z;

    const float* gate_base;
    const float* up_base;
    int rowbase;

    if (e < NEXP) {
        const int cnt = counts[e];
        if (chunk * 32 >= cnt) return;
        if (threadIdx.x < 32) {
            int i = chunk * 32 + (int)threadIdx.x;
            bool ok = (i < cnt);
            stok[threadIdx.x] = ok ? etok[e * NTOK + i] : -1;
            swt [threadIdx.x] = ok ? ewt [e * NTOK + i] : 0.f;
        }
        gate_base = w_gate_up + (size_t)e * (2 * FFN_DIM) * HID;
        up_base   = gate_base + (size_t)FFN_DIM * HID;
        rowbase   = offsets[e] + chunk * 32;
    } else {
        const int p = e - NEXP;            // shared pseudo-expert 0/1
        if (threadIdx.x < 32) {
            stok[threadIdx.x] = chunk * 32 + (int)threadIdx.x;
            swt [threadIdx.x] = 1.f;
        }
        gate_base = ws_gate_up + (size_t)(p * FFN_DIM) * HID;
        up_base   = ws_gate_up + (size_t)(SFFN + p * FFN_DIM) * HID;
        rowbase   = ROUTED_ROWS + p * NTOK + chunk * 32;
    }
    __syncthreads();

    const int wv   = threadIdx.x >> 5;
    const int lane = threadIdx.x & 31;
    const int half = lane >> 4;
    const int lq   = lane & 15;

    const int t0 = stok[lq];
    const int t1 = stok[16 + lq];
    const float* xrow0 = x + (size_t)((t0 < 0) ? 0 : t0) * HID + 2 * half;
    const float* xrow1 = x + (size_t)((t1 < 0) ? 0 : t1) * HID + 2 * half;

    const int f0 = (fsplit * 8 + wv) * 16;
    const float* gptr = gate_base + (size_t)(f0 + lq) * HID + 2 * half;
    const float* uptr = up_base   + (size_t)(f0 + lq) * HID + 2 * half;

    v8f accg0 = {}, accu0 = {}, accg1 = {}, accu1 = {};
    #pragma unroll 4
    for (int k = 0; k < HID; k += 4) {
        v2f a0 = *(const v2f*)(xrow0 + k);                          // L2-hot
        v2f a1 = *(const v2f*)(xrow1 + k);
        v2f bg = __builtin_nontemporal_load((const v2f*)(gptr + k)); // stream
        v2f bu = __builtin_nontemporal_load((const v2f*)(uptr + k)); // stream
        accg0 = wmma_f32_4(a0, bg, accg0);
        accu0 = wmma_f32_4(a0, bu, accu0);
        accg1 = wmma_f32_4(a1, bg, accg1);
        accu1 = wmma_f32_4(a1, bu, accu1);
    }

    float* arow = actws + (size_t)rowbase * FFN_DIM + f0 + lq;
    #pragma unroll
    for (int r = 0; r < 8; ++r) {
        const int m0 = r + 8 * half;
        const int m1 = m0 + 16;
        const float v0 = silu_fast(accg0[r]) * accu0[r] * swt[m0];
        const float v1 = silu_fast(accg1[r]) * accu1[r] * swt[m1];
        arow[(size_t)m0 * FFN_DIM] = v0;
        arow[(size_t)m1 * FFN_DIM] = v1;
    }
}

// ---------------------------------------------------------------------------
// Kernel 3: down GEMM, M=32 (w_down fragment reused across two M-tiles),
// atomic-accumulate into zeroed out.
// grid = (66 experts, 8 chunks, 16 h-splits of 128); block = 8 waves.
// ---------------------------------------------------------------------------
__global__ __launch_bounds__(256) void moe_down_kernel(
    const float* __restrict__ actws,       // [4096][1408]
    const float* __restrict__ w_down,      // [64][2048][1408]
    const float* __restrict__ ws_down,     // [2048][2816]
    const int*   __restrict__ counts,
    const int*   __restrict__ etok,
    const int*   __restrict__ offsets,
    float* __restrict__ out)               // [256][2048], pre-zeroed
{
    __shared__ int stok[32];

    const int e      = blockIdx.x;
    const int chunk  = blockIdx.y;
    const int hsplit = blockIdx.z;

    const float* down_base;
    int down_stride;
    int rowbase;

    if (e < NEXP) {
        const int cnt = counts[e];
        if (chunk * 32 >= cnt) return;
        if (threadIdx.x < 32) {
            int i = chunk * 32 + (int)threadIdx.x;
            stok[threadIdx.x] = (i < cnt) ? etok[e * NTOK + i] : -1;
        }
        down_base   = w_down + (size_t)e * HID * FFN_DIM;
        down_stride = FFN_DIM;
        rowbase     = offsets[e] + chunk * 32;
    } else {
        const int p = e - NEXP;
        if (threadIdx.x < 32) stok[threadIdx.x] = chunk * 32 + (int)threadIdx.x;
        down_base   = ws_down + (size_t)(p * FFN_DIM);
        down_stride = SFFN;
        rowbase     = ROUTED_ROWS + p * NTOK + chunk * 32;
    }
    __syncthreads();

    const int wv   = threadIdx.x >> 5;
    const int lane = threadIdx.x & 31;
    const int half = lane >> 4;
    const int lq   = lane & 15;

    const float* arow0 = actws + (size_t)(rowbase + lq) * FFN_DIM + 2 * half;
    const float* arow1 = arow0 + (size_t)16 * FFN_DIM;

    const int h0 = (hsplit * 8 + wv) * 16;
    const float* dptr = down_base + (size_t)(h0 + lq) * down_stride + 2 * half;

    v8f acc0 = {}, acc1 = {};
    #pragma unroll 4
    for (int k = 0; k < FFN_DIM; k += 4) {
        v2f a0 = *(const v2f*)(arow0 + k);                          // L2-hot
        v2f a1 = *(const v2f*)(arow1 + k);
        v2f b  = __builtin_nontemporal_load((const v2f*)(dptr + k)); // stream
        acc0 = wmma_f32_4(a0, b, acc0);
        acc1 = wmma_f32_4(a1, b, acc1);
    }

    #pragma unroll
    for (int r = 0; r < 8; ++r) {
        const int m0 = r + 8 * half;
        const int t0 = stok[m0];
        if (t0 >= 0) atomicAdd(&out[(size_t)t0 * HID + h0 + lq], acc0[r]);
        const int t1 = stok[m0 + 16];
        if (t1 >= 0) atomicAdd(&out[(size_t)t1 * HID + h0 + lq], acc1[r]);
    }
}

// ---------------------------------------------------------------------------
// Fallback (small ws): fused M=16 kernel with act tile in LDS (round-1 design)
// ---------------------------------------------------------------------------
__global__ __launch_bounds__(256) void moe_fused_kernel(
    const float* __restrict__ x,
    const float* __restrict__ w_gate_up,
    const float* __restrict__ w_down,
    const float* __restrict__ ws_gate_up,
    const float* __restrict__ ws_down,
    const int*   __restrict__ counts,
    const int*   __restrict__ etok,
    const float* __restrict__ ewt,
    float* __restrict__ out)
{
    __shared__ float act[16 * LDS_STR];
    __shared__ int   stok[16];
    __shared__ float swt[16];

    const int e     = blockIdx.x;
    const int chunk = blockIdx.y;

    const float* gate_base;
    const float* up_base;
    const float* down_base;
    int down_stride;

    if (e < NEXP) {
        const int cnt = counts[e];
        if (chunk * 16 >= cnt) return;
        if (threadIdx.x < 16) {
            int i = chunk * 16 + (int)threadIdx.x;
            bool ok = (i < cnt);
            stok[threadIdx.x] = ok ? etok[e * NTOK + i] : -1;
            swt [threadIdx.x] = ok ? ewt [e * NTOK + i] : 0.f;
        }
        gate_base   = w_gate_up + (size_t)e * (2 * FFN_DIM) * HID;
        up_base     = gate_base + (size_t)FFN_DIM * HID;
        down_base   = w_down + (size_t)e * HID * FFN_DIM;
        down_stride = FFN_DIM;
    } else {
        const int p = e - NEXP;
        if (threadIdx.x < 16) {
            stok[threadIdx.x] = chunk * 16 + (int)threadIdx.x;
            swt [threadIdx.x] = 1.f;
        }
        gate_base   = ws_gate_up + (size_t)(p * FFN_DIM) * HID;
        up_base     = ws_gate_up + (size_t)(SFFN + p * FFN_DIM) * HID;
        down_base   = ws_down + (size_t)(p * FFN_DIM);
        down_stride = SFFN;
    }
    __syncthreads();

    const int wv   = threadIdx.x >> 5;
    const int lane = threadIdx.x & 31;
    const int half = lane >> 4;
    const int lq   = lane & 15;

    const int tA = stok[lq];
    const float* xrow = x + (size_t)((tA < 0) ? 0 : tA) * HID + 2 * half;

    for (int ft = wv; ft < FFN_DIM / 16; ft += 8) {
        const int f0 = ft * 16;
        const float* gptr = gate_base + (size_t)(f0 + lq) * HID + 2 * half;
        const float* uptr = up_base   + (size_t)(f0 + lq) * HID + 2 * half;
        v8f accg = {}, accu = {};
        #pragma unroll 4
        for (int k = 0; k < HID; k += 4) {
            v2f a  = *(const v2f*)(xrow + k);
            v2f bg = __builtin_nontemporal_load((const v2f*)(gptr + k));
            v2f bu = __builtin_nontemporal_load((const v2f*)(uptr + k));
            accg = wmma_f32_4(a, bg, accg);
            accu = wmma_f32_4(a, bu, accu);
        }
        #pragma unroll
        for (int r = 0; r < 8; ++r) {
            const int m = r + 8 * half;
            act[m * LDS_STR + f0 + lq] = silu_fast(accg[r]) * accu[r] * swt[m];
        }
    }
    __syncthreads();

    const float* aLds = &act[lq * LDS_STR + 2 * half];
    for (int ht = wv; ht < HID / 16; ht += 8) {
        const int h0 = ht * 16;
        const float* dptr = down_base + (size_t)(h0 + lq) * down_stride + 2 * half;
        v8f acc = {};
        #pragma unroll 4
        for (int k = 0; k < FFN_DIM; k += 4) {
            v2f a = *(const v2f*)(aLds + k);
            v2f b = __builtin_nontemporal_load((const v2f*)(dptr + k));
            acc = wmma_f32_4(a, b, acc);
        }
        #pragma unroll
        for (int r = 0; r < 8; ++r) {
            const int m = r + 8 * half;
            const int t = stok[m];
            if (t >= 0) atomicAdd(&out[(size_t)t * HID + h0 + lq], acc[r]);
        }
    }
}

// ---------------------------------------------------------------------------
// Host launcher
// ---------------------------------------------------------------------------
extern "C" void kernel_launch(void* const* d_in, const int* in_sizes, int n_in,
                              void* d_out, int out_size, void* d_ws, size_t ws_size,
                              hipStream_t stream) {
    const float* x          = (const float*)d_in[0]; // (1,256,2048)
    const float* gate_w     = (const float*)d_in[1]; // (64,2048)
    const float* w_gate_up  = (const float*)d_in[2]; // (64,2816,2048)
    const float* w_down     = (const float*)d_in[3]; // (64,2048,1408)
    const float* ws_gate_up = (const float*)d_in[4]; // (5632,2048)
    const float* ws_down    = (const float*)d_in[5]; // (2048,2816)
    float* out = (float*)d_out;

    // ws layout: counts[64] | etok[64*256] | ewt[64*256] | offsets[64] | act
    int*   counts  = (int*)d_ws;
    int*   etok    = counts + NEXP;
    float* ewt     = (float*)(etok + NEXP * NTOK);
    int*   offsets = (int*)(ewt + NEXP * NTOK);
    float* actws   = (float*)(offsets + NEXP);

    const size_t head  = (size_t)(NEXP + NEXP * NTOK + NEXP * NTOK + NEXP) * 4;
    const size_t need  = head + (size_t)ACT_ROWS * FFN_DIM * sizeof(float);

    hipMemsetAsync(counts, 0, NEXP * sizeof(int), stream);
    hipMemsetAsync(out, 0, (size_t)out_size * sizeof(float), stream);

    moe_route_kernel<<<NTOK, 64, 0, stream>>>(x, gate_w, counts, etok, ewt);

    if (ws_size >= need) {
        moe_offsets_kernel<<<1, 64, 0, stream>>>(counts, offsets);
        dim3 gA(NEXP + 2, 8, FFN_DIM / 128);   // 66 x 8 x 11
        moe_gateup_kernel<<<gA, 256, 0, stream>>>(
            x, w_gate_up, ws_gate_up, counts, etok, ewt, offsets, actws);
        dim3 gB(NEXP + 2, 8, HID / 128);       // 66 x 8 x 16
        moe_down_kernel<<<gB, 256, 0, stream>>>(
            actws, w_down, ws_down, counts, etok, offsets, out);
    } else {
        dim3 grid(NEXP + 2, 16);
        moe_fused_kernel<<<grid, 256, 0, stream>>>(
            x, w_gate_up, w_down, ws_gate_up, ws_down, counts, etok, ewt, out);
    }
}